// MultiLevelGraphLayer_9594956939316
// MI455X (gfx1250) — compile-verified
//
#include <hip/hip_runtime.h>
#include <hip/hip_bf16.h>

#define N_HIGH 4096
#define N_LOW  262144
#define E_HIGH 65536
#define E_LOW  524288
#define DIM    128
#define HEADS  4
#define EPS    1e-5f
#define SCALE  0.17677669529663687f   // 1/sqrt(32)

typedef __attribute__((ext_vector_type(2))) float v2f;
typedef __attribute__((ext_vector_type(8))) float v8f;

// ---- monotone uint encoding of float for atomicMax-based segment_max ----
__device__ __forceinline__ unsigned enc_f(float x) {
    unsigned b = __float_as_uint(x);
    return (b & 0x80000000u) ? ~b : (b | 0x80000000u);
}
__device__ __forceinline__ float dec_f(unsigned u) {
    unsigned b = (u & 0x80000000u) ? (u & 0x7fffffffu) : ~u;
    return __uint_as_float(b);
}

__device__ __forceinline__ void atomic_add_f(float* p, float v) {
    __hip_atomic_fetch_add(p, v, __ATOMIC_RELAXED, __HIP_MEMORY_SCOPE_AGENT);
}

// =====================================================================
// Fused Q/K/V/skip projection: out = x @ W + b via V_WMMA_F32_16X16X4_F32
// block = 128 threads = 4 waves; wave w handles weight matrix w.
// grid = (N/16 row-tiles, 8 column-tiles of 16)
// =====================================================================
__global__ __launch_bounds__(128) void qkvs_gemm(
    const float* __restrict__ x,
    const float* __restrict__ Wq, const float* __restrict__ bq,
    const float* __restrict__ Wk, const float* __restrict__ bk,
    const float* __restrict__ Wv, const float* __restrict__ bv,
    const float* __restrict__ Ws, const float* __restrict__ bs,
    float* __restrict__ outQ, float* __restrict__ outK,
    float* __restrict__ outV, float* __restrict__ outA)
{
    __shared__ float xs[16 * 132];            // padded stride: bank-conflict-free
    const int tid     = threadIdx.x;
    const int rowBase = blockIdx.x * 16;
    const int colTile = blockIdx.y;           // 0..7

    // cooperative load of the 16x128 A tile (coalesced)
    #pragma unroll
    for (int i = 0; i < 16; ++i) {
        int idx = tid + i * 128;
        int r = idx >> 7, c = idx & 127;
        xs[r * 132 + c] = x[(size_t)(rowBase + r) * DIM + c];
    }
    __syncthreads();

    const int wave = tid >> 5;
    const int lane = tid & 31;
    const float* W; const float* bias; float* out;
    if      (wave == 0) { W = Wq; bias = bq; out = outQ; }
    else if (wave == 1) { W = Wk; bias = bk; out = outK; }
    else if (wave == 2) { W = Wv; bias = bv; out = outV; }
    else                { W = Ws; bias = bs; out = outA; }

    const int m   = lane & 15;                // A: M index
    const int kh  = (lane >> 4) << 1;         // A/B: K sub-offset (0 or 2)
    const int col = colTile * 16 + (lane & 15);

    v8f acc = {0.f, 0.f, 0.f, 0.f, 0.f, 0.f, 0.f, 0.f};
    #pragma unroll
    for (int k0 = 0; k0 < DIM; k0 += 4) {
        v2f A, B;
        A[0] = xs[m * 132 + k0 + kh];
        A[1] = xs[m * 132 + k0 + kh + 1];
        B[0] = W[(size_t)(k0 + kh)     * DIM + col];
        B[1] = W[(size_t)(k0 + kh + 1) * DIM + col];
        acc = __builtin_amdgcn_wmma_f32_16x16x4_f32(
            false, A, false, B, (short)0, acc, false, false);
    }
    const float bb   = bias[col];
    const int   rOff = (lane >> 4) << 3;      // C/D: M = r + 8*(lane>=16)
    #pragma unroll
    for (int r = 0; r < 8; ++r)
        out[(size_t)(rowBase + rOff + r) * DIM + col] = acc[r] + bb;
}

// =====================================================================
// Warp per edge: per-head score = (q[dst] . k[src]) * SCALE, atomic max
// =====================================================================
__global__ __launch_bounds__(256) void edge_scores(
    const float* __restrict__ Q, const float* __restrict__ K,
    const int* __restrict__ eidx, int E,
    float* __restrict__ score, unsigned* __restrict__ nmax)
{
    int e = blockIdx.x * 8 + (threadIdx.x >> 5);
    if (e >= E) return;
    int lane = threadIdx.x & 31;
    int src = eidx[e];
    int dst = eidx[E + e];
    const float* q = Q + (size_t)dst * DIM;
    const float* k = K + (size_t)src * DIM;
    float s[4];
    #pragma unroll
    for (int j = 0; j < 4; ++j) s[j] = q[lane + 32 * j] * k[lane + 32 * j];
    #pragma unroll
    for (int off = 16; off > 0; off >>= 1) {
        #pragma unroll
        for (int j = 0; j < 4; ++j) s[j] += __shfl_xor(s[j], off, 32);
    }
    if (lane < 4) {
        float v = s[lane] * SCALE;
        score[(size_t)e * 4 + lane] = v;
        atomicMax(&nmax[(size_t)dst * 4 + lane], enc_f(v));
    }
}

// thread per (edge, head): e = exp(score - max), denom += e
__global__ __launch_bounds__(256) void edge_exp(
    const int* __restrict__ eidx, int E,
    float* __restrict__ score, const unsigned* __restrict__ nmax,
    float* __restrict__ denom)
{
    int idx = blockIdx.x * 256 + threadIdx.x;
    if (idx >= E * 4) return;
    int e = idx >> 2, h = idx & 3;
    int dst = eidx[E + e];
    float m  = dec_f(nmax[(size_t)dst * 4 + h]);
    float ex = __expf(score[idx] - m);
    score[idx] = ex;
    atomic_add_f(&denom[(size_t)dst * 4 + h], ex);
}

// warp per edge: acc[dst] += (e/denom) * v[src]  (acc pre-seeded with skip)
__global__ __launch_bounds__(256) void edge_agg(
    const int* __restrict__ eidx, int E,
    const float* __restrict__ escore, const float* __restrict__ denom,
    const float* __restrict__ V, float* __restrict__ acc)
{
    int e = blockIdx.x * 8 + (threadIdx.x >> 5);
    if (e >= E) return;
    int lane = threadIdx.x & 31;
    int src = eidx[e];
    int dst = eidx[E + e];
    #pragma unroll
    for (int j = 0; j < 4; ++j) {
        float alpha = escore[(size_t)e * 4 + j] / denom[(size_t)dst * 4 + j];
        int c = lane + 32 * j;
        atomic_add_f(&acc[(size_t)dst * DIM + c], alpha * V[(size_t)src * DIM + c]);
    }
}

// warp per node: ReLU(LayerNorm(x))
__global__ __launch_bounds__(256) void ln_relu(
    const float* __restrict__ in, const float* __restrict__ w,
    const float* __restrict__ b, float* __restrict__ out, int N)
{
    int n = blockIdx.x * 8 + (threadIdx.x >> 5);
    if (n >= N) return;
    int lane = threadIdx.x & 31;
    float x[4], sum = 0.f, sq = 0.f;
    #pragma unroll
    for (int j = 0; j < 4; ++j) {
        x[j] = in[(size_t)n * DIM + lane + 32 * j];
        sum += x[j]; sq += x[j] * x[j];
    }
    #pragma unroll
    for (int off = 16; off > 0; off >>= 1) {
        sum += __shfl_xor(sum, off, 32);
        sq  += __shfl_xor(sq,  off, 32);
    }
    float mean = sum * (1.f / 128.f);
    float var  = sq  * (1.f / 128.f) - mean * mean;
    float inv  = rsqrtf(var + EPS);
    #pragma unroll
    for (int j = 0; j < 4; ++j) {
        int c = lane + 32 * j;
        float y = (x[j] - mean) * inv * w[c] + b[c];
        out[(size_t)n * DIM + c] = fmaxf(y, 0.f);
    }
}

// warp per low node: pooled[batch[n]] += l[n]; counts[batch[n]] += 1
__global__ __launch_bounds__(256) void pool_sum(
    const float* __restrict__ l, const int* __restrict__ batch,
    float* __restrict__ pooled, float* __restrict__ counts)
{
    int n = blockIdx.x * 8 + (threadIdx.x >> 5);
    if (n >= N_LOW) return;
    int lane = threadIdx.x & 31;
    int g = batch[n];
    #pragma unroll
    for (int j = 0; j < 4; ++j) {
        int c = lane + 32 * j;
        atomic_add_f(&pooled[(size_t)g * DIM + c], l[(size_t)n * DIM + c]);
    }
    if (lane == 0) atomic_add_f(&counts[g], 1.f);
}

__global__ __launch_bounds__(256) void pool_div(
    float* __restrict__ pooled, const float* __restrict__ counts)
{
    int idx = blockIdx.x * 256 + threadIdx.x;
    if (idx >= N_HIGH * DIM) return;
    pooled[idx] /= fmaxf(counts[idx >> 7], 1.f);
}

// warp per high node: sim = sigmoid([h,pooled]@w_lh); out = sim*h + (1-sim)*pooled
__global__ __launch_bounds__(256) void combine_high(
    const float* __restrict__ h, const float* __restrict__ pooled,
    const float* __restrict__ w_lh, float* __restrict__ outH)
{
    int n = blockIdx.x * 8 + (threadIdx.x >> 5);
    if (n >= N_HIGH) return;
    int lane = threadIdx.x & 31;
    float hv[4], pv[4], dot = 0.f;
    #pragma unroll
    for (int j = 0; j < 4; ++j) {
        int c = lane + 32 * j;
        hv[j] = h[(size_t)n * DIM + c];
        pv[j] = pooled[(size_t)n * DIM + c];
        dot += hv[j] * w_lh[c] + pv[j] * w_lh[DIM + c];
    }
    #pragma unroll
    for (int off = 16; off > 0; off >>= 1) dot += __shfl_xor(dot, off, 32);
    float sim = 1.f / (1.f + __expf(-dot));
    #pragma unroll
    for (int j = 0; j < 4; ++j) {
        int c = lane + 32 * j;
        outH[(size_t)n * DIM + c] = sim * hv[j] + (1.f - sim) * pv[j];
    }
}

// warp per low node (in-place on outLow): a = sigmoid([h[batch],l]@w_hl)
__global__ __launch_bounds__(256) void combine_low(
    const float* __restrict__ hN, const int* __restrict__ batch,
    const float* __restrict__ w_hl, float* __restrict__ outL)
{
    int n = blockIdx.x * 8 + (threadIdx.x >> 5);
    if (n >= N_LOW) return;
    int lane = threadIdx.x & 31;
    int g = batch[n];
    float lv[4], hb[4], dot = 0.f;
    #pragma unroll
    for (int j = 0; j < 4; ++j) {
        int c = lane + 32 * j;
        lv[j] = outL[(size_t)n * DIM + c];
        hb[j] = hN[(size_t)g * DIM + c];
        dot += hb[j] * w_hl[c] + lv[j] * w_hl[DIM + c];
    }
    #pragma unroll
    for (int off = 16; off > 0; off >>= 1) dot += __shfl_xor(dot, off, 32);
    float a = 1.f / (1.f + __expf(-dot));
    #pragma unroll
    for (int j = 0; j < 4; ++j) {
        int c = lane + 32 * j;
        outL[(size_t)n * DIM + c] = a * lv[j] + (1.f - a) * hb[j];
    }
}

extern "C" void kernel_launch(void* const* d_in, const int* in_sizes, int n_in,
                              void* d_out, int out_size, void* d_ws, size_t ws_size,
                              hipStream_t stream) {
    const float* high_emb = (const float*)d_in[0];
    const float* low_emb  = (const float*)d_in[1];
    const float* Wq = (const float*)d_in[2];  const float* bq = (const float*)d_in[3];
    const float* Wk = (const float*)d_in[4];  const float* bk = (const float*)d_in[5];
    const float* Wv = (const float*)d_in[6];  const float* bv = (const float*)d_in[7];
    const float* Ws = (const float*)d_in[8];  const float* bs = (const float*)d_in[9];
    const float* ln_w = (const float*)d_in[10];
    const float* ln_b = (const float*)d_in[11];
    const float* w_lh = (const float*)d_in[12];
    const float* w_hl = (const float*)d_in[13];
    const int* eH    = (const int*)d_in[14];
    const int* eL    = (const int*)d_in[15];
    const int* batch = (const int*)d_in[16];

    float* outHigh = (float*)d_out;
    float* outLow  = outHigh + (size_t)N_HIGH * DIM;

    // ---- workspace carve-out ----
    char* p = (char*)d_ws;
    auto alloc = [&](size_t nf) { float* r = (float*)p; p += nf * sizeof(float); return r; };
    float* lowQ  = alloc((size_t)N_LOW * DIM);
    float* lowK  = alloc((size_t)N_LOW * DIM);
    float* lowV  = alloc((size_t)N_LOW * DIM);
    float* lowA  = alloc((size_t)N_LOW * DIM);
    float* highQ = alloc((size_t)N_HIGH * DIM);
    float* highK = alloc((size_t)N_HIGH * DIM);
    float* highV = alloc((size_t)N_HIGH * DIM);
    float* highA = alloc((size_t)N_HIGH * DIM);
    float* highN = alloc((size_t)N_HIGH * DIM);   // normalized h
    float* pooled = alloc((size_t)N_HIGH * DIM);
    float* counts = alloc(N_HIGH);
    float* scoreL = alloc((size_t)E_LOW * HEADS);
    float* scoreH = alloc((size_t)E_HIGH * HEADS);
    float* denomL = alloc((size_t)N_LOW * HEADS);
    float* denomH = alloc((size_t)N_HIGH * HEADS);
    unsigned* maxL = (unsigned*)alloc((size_t)N_LOW * HEADS);
    unsigned* maxH = (unsigned*)alloc((size_t)N_HIGH * HEADS);

    // ---- zero/neutral init (graph-capturable memsets) ----
    hipMemsetAsync(maxL,   0, (size_t)N_LOW  * HEADS * 4, stream);
    hipMemsetAsync(maxH,   0, (size_t)N_HIGH * HEADS * 4, stream);
    hipMemsetAsync(denomL, 0, (size_t)N_LOW  * HEADS * 4, stream);
    hipMemsetAsync(denomH, 0, (size_t)N_HIGH * HEADS * 4, stream);
    hipMemsetAsync(pooled, 0, (size_t)N_HIGH * DIM * 4, stream);
    hipMemsetAsync(counts, 0, (size_t)N_HIGH * 4, stream);

    // ---- projections (WMMA) ----
    qkvs_gemm<<<dim3(N_HIGH / 16, 8), 128, 0, stream>>>(
        high_emb, Wq, bq, Wk, bk, Wv, bv, Ws, bs, highQ, highK, highV, highA);
    qkvs_gemm<<<dim3(N_LOW / 16, 8), 128, 0, stream>>>(
        low_emb, Wq, bq, Wk, bk, Wv, bv, Ws, bs, lowQ, lowK, lowV, lowA);

    // ---- attention (scores -> softmax -> aggregate) ----
    edge_scores<<<E_HIGH / 8, 256, 0, stream>>>(highQ, highK, eH, E_HIGH, scoreH, maxH);
    edge_scores<<<E_LOW  / 8, 256, 0, stream>>>(lowQ,  lowK,  eL, E_LOW,  scoreL, maxL);
    edge_exp<<<(E_HIGH * 4) / 256, 256, 0, stream>>>(eH, E_HIGH, scoreH, maxH, denomH);
    edge_exp<<<(E_LOW  * 4) / 256, 256, 0, stream>>>(eL, E_LOW,  scoreL, maxL, denomL);
    edge_agg<<<E_HIGH / 8, 256, 0, stream>>>(eH, E_HIGH, scoreH, denomH, highV, highA);
    edge_agg<<<E_LOW  / 8, 256, 0, stream>>>(eL, E_LOW,  scoreL, denomL, lowV, lowA);

    // ---- LayerNorm + ReLU ----
    ln_relu<<<N_HIGH / 8, 256, 0, stream>>>(highA, ln_w, ln_b, highN, N_HIGH);
    ln_relu<<<N_LOW  / 8, 256, 0, stream>>>(lowA,  ln_w, ln_b, outLow, N_LOW);

    // ---- pooling ----
    pool_sum<<<N_LOW / 8, 256, 0, stream>>>(outLow, batch, pooled, counts);
    pool_div<<<(N_HIGH * DIM) / 256, 256, 0, stream>>>(pooled, counts);

    // ---- gated combines ----
    combine_high<<<N_HIGH / 8, 256, 0, stream>>>(highN, pooled, w_lh, outHigh);
    combine_low<<<N_LOW  / 8, 256, 0, stream>>>(highN, batch, w_hl, outLow);
}